// JointEdgeDiceLoss_90795608638213
// MI455X (gfx1250) — compile-verified
//
#include <hip/hip_runtime.h>
#include <hip/hip_bf16.h>

typedef float v2f __attribute__((ext_vector_type(2)));
typedef float v8f __attribute__((ext_vector_type(8)));

#define N_   2
#define C_   4
#define S_   (96 * 160 * 160)   // 2,457,600 voxels per (n, c) plane
#define S4_  (S_ / 4)           // float4 groups per batch item
#define V_   (N_ * S_)          // total voxels = 4,915,200
#define V4_  (V_ / 4)           // total float4 groups
#define BLK  256
#define PITCH 17                // LDS row pitch (floats), conflict-free
#define SMOOTH 1e-5f

// ---------------------------------------------------------------------------
// Pass 1: streaming kernel. Each thread accumulates 16 partial sums:
//   [0..3]  intersect[c]   = sum probs[c] where target==c
//   [4..7]  probsum[c]     = sum probs[c]
//   [8..11] count[c]       = #voxels with target==c
//   [12]    sum bce where edge target==1
//   [13]    sum bce where edge target==0
//   [14]    pos_num, [15] neg_num
// Block reduction: stage 256x16 accumulators in LDS, then wave 0 chains
// 64 x V_WMMA_F32_16X16X4_F32 with a ones A-matrix (full-f32 column sums).
// ---------------------------------------------------------------------------
__global__ __launch_bounds__(BLK) void fused_partials_kernel(
    const float* __restrict__ segin,
    const int*   __restrict__ segmask,
    const float* __restrict__ edgein,
    const int*   __restrict__ edgemask,
    float*       __restrict__ partials)
{
  float acc[16];
#pragma unroll
  for (int j = 0; j < 16; ++j) acc[j] = 0.f;

  const float L2E = 1.4426950408889634f;   // log2(e)
  const float LN2 = 0.6931471805599453f;   // ln(2)

  unsigned stride = gridDim.x * BLK;
  for (unsigned g = blockIdx.x * BLK + threadIdx.x; g < V4_; g += stride) {
    unsigned nb = g / S4_;                  // batch index
    unsigned s  = (g - nb * S4_) * 4u;      // voxel offset within plane
    const float* base = segin + (size_t)nb * (C_ * S_) + s;
    float4 r0 = *(const float4*)(base);
    float4 r1 = *(const float4*)(base + S_);
    float4 r2 = *(const float4*)(base + 2 * S_);
    float4 r3 = *(const float4*)(base + 3 * S_);
    size_t mi = (size_t)nb * S_ + s;
    int4   tm = *(const int4*)(segmask + mi);
    float4 ex = *(const float4*)(edgein + mi);
    int4   em = *(const int4*)(edgemask + mi);

    const float* f0 = (const float*)&r0;
    const float* f1 = (const float*)&r1;
    const float* f2 = (const float*)&r2;
    const float* f3 = (const float*)&r3;
    const int*   ti = (const int*)&tm;
    const float* ef = (const float*)&ex;
    const int*   ei = (const int*)&em;

#pragma unroll
    for (int v = 0; v < 4; ++v) {
      // ---- softmax over C=4 + dice partials ----
      float x0 = f0[v], x1 = f1[v], x2 = f2[v], x3 = f3[v];
      float m  = fmaxf(fmaxf(x0, x1), fmaxf(x2, x3));
      float e0 = __builtin_amdgcn_exp2f((x0 - m) * L2E);
      float e1 = __builtin_amdgcn_exp2f((x1 - m) * L2E);
      float e2 = __builtin_amdgcn_exp2f((x2 - m) * L2E);
      float e3 = __builtin_amdgcn_exp2f((x3 - m) * L2E);
      float inv = __builtin_amdgcn_rcpf(e0 + e1 + e2 + e3);
      float p0 = e0 * inv, p1 = e1 * inv, p2 = e2 * inv, p3 = e3 * inv;
      acc[4] += p0; acc[5] += p1; acc[6] += p2; acc[7] += p3;
      int t = ti[v];
      float s0 = (t == 0) ? 1.f : 0.f;
      float s1 = (t == 1) ? 1.f : 0.f;
      float s2 = (t == 2) ? 1.f : 0.f;
      float s3 = (t == 3) ? 1.f : 0.f;
      acc[0] += s0 * p0; acc[1] += s1 * p1; acc[2] += s2 * p2; acc[3] += s3 * p3;
      acc[8] += s0;      acc[9] += s1;      acc[10] += s2;     acc[11] += s3;

      // ---- weighted-BCE partials ----
      float x  = ef[v];
      int   tb = ei[v];
      float ax = fabsf(x);
      float z  = __builtin_amdgcn_exp2f(-ax * L2E);
      float sp = __builtin_amdgcn_logf(1.f + z) * LN2;   // log1p(exp(-|x|))
      float core = fmaxf(x, 0.f) - x * (float)tb + sp;
      float bp = (tb == 1) ? 1.f : 0.f;
      float bn = (tb == 0) ? 1.f : 0.f;
      acc[12] += bp * core;
      acc[13] += bn * core;
      acc[14] += bp;
      acc[15] += bn;
    }
  }

  // ---- block reduction: 256 threads x 16 accs -> 16 sums via WMMA ----
  __shared__ float lds[BLK * PITCH];
#pragma unroll
  for (int j = 0; j < 16; ++j) lds[threadIdx.x * PITCH + j] = acc[j];
  __syncthreads();

  if (threadIdx.x < 32) {              // wave 0 only; EXEC all-ones inside
    int lane = threadIdx.x;
    int col  = lane & 15;
    int half = lane >> 4;
    v2f a; a[0] = 1.f; a[1] = 1.f;     // A = ones(16x4), layout-agnostic
    v8f c = {};
#pragma unroll 4
    for (int it = 0; it < BLK / 4; ++it) {
      int r = it * 4;                  // 4 thread-rows per WMMA
      v2f b;
      b[0] = lds[(r + half)     * PITCH + col];
      b[1] = lds[(r + 2 + half) * PITCH + col];
      // D = ones(16x4) * B(4x16) + C : every D row = column sums of B
      c = __builtin_amdgcn_wmma_f32_16x16x4_f32(false, a, false, b,
                                                (short)0, c, false, false);
    }
    if (lane < 16)                     // D[0][n] lives in VGPR0, lanes 0-15
      partials[(size_t)blockIdx.x * 16 + lane] = c[0];
  }
}

// ---------------------------------------------------------------------------
// Pass 2: reduce per-block partials and emit (region_loss, edge_loss).
// ---------------------------------------------------------------------------
__global__ __launch_bounds__(BLK) void finalize_kernel(
    const float* __restrict__ partials, int nblocks, float* __restrict__ out)
{
  int j = threadIdx.x & 15;
  int s = threadIdx.x >> 4;
  float t = 0.f;
  for (int b = s; b < nblocks; b += 16) t += partials[(size_t)b * 16 + j];
  __shared__ float m[16][17];
  __shared__ float totals[16];
  m[s][j] = t;
  __syncthreads();
  if (threadIdx.x < 16) {
    float tot = 0.f;
#pragma unroll
    for (int ss = 0; ss < 16; ++ss) tot += m[ss][threadIdx.x];
    totals[threadIdx.x] = tot;
  }
  __syncthreads();
  if (threadIdx.x == 0) {
    float dsum = 0.f;
#pragma unroll
    for (int c = 0; c < 4; ++c) {
      float I = totals[c], P = totals[4 + c], K = totals[8 + c];
      dsum += (2.f * I + SMOOTH) / (P + K + SMOOTH);
    }
    out[0] = 1.f - 0.25f * dsum;                 // region (dice) loss
    float pos = totals[14], neg = totals[15];
    float inv_s = 1.f / (pos + neg);
    out[1] = (neg * totals[12] + pos * totals[13]) * inv_s
             * (1.f / (float)V_);                // edge (weighted BCE) loss
  }
}

extern "C" void kernel_launch(void* const* d_in, const int* in_sizes, int n_in,
                              void* d_out, int out_size, void* d_ws, size_t ws_size,
                              hipStream_t stream) {
  const float* segin    = (const float*)d_in[0];
  const float* edgein   = (const float*)d_in[1];
  const int*   segmask  = (const int*)d_in[2];
  const int*   edgemask = (const int*)d_in[3];
  float* out      = (float*)d_out;
  float* partials = (float*)d_ws;

  size_t maxb = ws_size / (16 * sizeof(float));
  int nblocks = 2048;                       // ~512K threads: saturates HBM
  if ((size_t)nblocks > maxb) nblocks = (int)maxb;
  if (nblocks < 1) nblocks = 1;

  fused_partials_kernel<<<nblocks, BLK, 0, stream>>>(segin, segmask, edgein,
                                                     edgemask, partials);
  finalize_kernel<<<1, BLK, 0, stream>>>(partials, nblocks, out);
}